// SegmentedPolynomialFusedTP_85779086836289
// MI455X (gfx1250) — compile-verified
//
#include <hip/hip_runtime.h>

// Segmented TP 'u_u_u': out[dst[e]] += x[src[e]] * y[e] * repeat(coeffs, 32)
// E = 1e6, D = 128 (4 segments of 32). Memory-bound: ~620 MB HBM -> ~27us @ 23.3 TB/s.
//
// CDNA5 path: the 512 MB y stream is moved with the async DMA path
// (global_load_async_to_lds_b128, ASYNCcnt) into a per-wave double buffer in
// LDS, prefetching the next edge's 512 B tile while the current edge's
// gather + scatter-atomics execute. x (51 MB) and out (51 MB) stay resident
// in the 192 MB L2; y is tagged non-temporal so it never evicts them.

#define TP_D            128
#define WAVES_PER_BLOCK 8
#define EDGE_BYTES      512            // 128 f32
#define LDS_PER_WAVE    (2 * EDGE_BYTES)

typedef float float4v __attribute__((ext_vector_type(4)));

__global__ __launch_bounds__(256)
void seg_tp_scatter_async(const float* __restrict__ x,
                          const float* __restrict__ y,
                          const float* __restrict__ coeffs,
                          const int*   __restrict__ src,
                          const int*   __restrict__ dst,
                          float*       __restrict__ out,
                          int n_edges)
{
    __shared__ __align__(16) unsigned char lds_buf[WAVES_PER_BLOCK * LDS_PER_WAVE];

    const int lane        = threadIdx.x & 31;
    const int waveInBlock = threadIdx.x >> 5;

    const long long wave0 = (long long)blockIdx.x * WAVES_PER_BLOCK + waveInBlock;
    const long long step  = (long long)gridDim.x * WAVES_PER_BLOCK;

    // Lane covers columns [4*lane, 4*lane+3]; all in segment lane/8.
    const float c   = coeffs[lane >> 3];
    const int   col = lane << 2;

    // Hardware LDS byte offset of this lane's 16B slot in buffer 0.
    // (low 32 bits of a generic shared-memory address == LDS offset)
    const unsigned ldsLane0 = (unsigned)(size_t)lds_buf
                            + (unsigned)(waveInBlock * LDS_PER_WAVE)
                            + (unsigned)(lane * 16);

    long long e = wave0;

    // Prologue: start async DMA of the first y tile into buffer 0.
    if (e < n_edges) {
        const float* gp = y + e * TP_D + col;
        asm volatile("global_load_async_to_lds_b128 %0, %1, off th:TH_LOAD_NT"
                     :: "v"(ldsLane0), "v"(gp) : "memory");
    }

    unsigned cur = 0;
    for (; e < n_edges; e += step) {
        const long long en      = e + step;
        const bool      hasNext = (en < n_edges);   // wave-uniform

        if (hasNext) {
            // Prefetch next tile into the other buffer, then wait until only
            // that newest copy is outstanding (in-order completion => current
            // buffer's DMA is done).
            const float* gp = y + en * TP_D + col;
            asm volatile("global_load_async_to_lds_b128 %0, %1, off th:TH_LOAD_NT"
                         :: "v"(ldsLane0 + ((cur ^ 1u) * (unsigned)EDGE_BYTES)),
                            "v"(gp) : "memory");
            asm volatile("s_wait_asynccnt 0x1" ::: "memory");
        } else {
            asm volatile("s_wait_asynccnt 0x0" ::: "memory");
        }

        // Wave-uniform indices -> SGPRs (GVS addressing for load/atomics).
        const int s = __builtin_amdgcn_readfirstlane(src[e]);
        const int d = __builtin_amdgcn_readfirstlane(dst[e]);

        // Read the DMA'd y tile back from LDS (ds_load_b128).
        const float4v yv = *(const float4v*)(lds_buf
                              + waveInBlock * LDS_PER_WAVE
                              + cur * EDGE_BYTES
                              + lane * 16);

        // x gather: hot in 192 MB L2.
        const float4v xv = *(const float4v*)(x + (long long)s * TP_D + col);

        float4v p;
        p.x = xv.x * yv.x * c;
        p.y = xv.y * yv.y * c;
        p.z = xv.z * yv.z * c;
        p.w = xv.w * yv.w * c;

        // Scatter-add via hardware global_atomic_add_f32 (no return).
        float* o = out + (long long)d * TP_D + col;
        unsafeAtomicAdd(o + 0, p.x);
        unsafeAtomicAdd(o + 1, p.y);
        unsafeAtomicAdd(o + 2, p.z);
        unsafeAtomicAdd(o + 3, p.w);

        cur ^= 1u;
    }
}

extern "C" void kernel_launch(void* const* d_in, const int* in_sizes, int n_in,
                              void* d_out, int out_size, void* d_ws, size_t ws_size,
                              hipStream_t stream) {
    const float* x      = (const float*)d_in[0];   // [N_NODES, 128] f32
    const float* y      = (const float*)d_in[1];   // [N_EDGES, 128] f32
    const float* coeffs = (const float*)d_in[2];   // [4] f32
    const int*   src    = (const int*)d_in[3];     // [N_EDGES]
    const int*   dst    = (const int*)d_in[4];     // [N_EDGES]
    float*       out    = (float*)d_out;           // [N_NODES, 128] f32

    const int n_edges = in_sizes[3];

    // Harness poisons d_out; the scatter-add needs zeros first.
    hipMemsetAsync(d_out, 0, (size_t)out_size * sizeof(float), stream);

    long long wavesNeeded = (long long)n_edges;
    int blocks = (int)((wavesNeeded + WAVES_PER_BLOCK - 1) / WAVES_PER_BLOCK);
    if (blocks > 16384) blocks = 16384;   // grid-stride: ~8 edges/wave
    if (blocks < 1) blocks = 1;

    seg_tp_scatter_async<<<blocks, 256, 0, stream>>>(
        x, y, coeffs, src, dst, out, n_edges);
}